// GraphConvolution_30880814858345
// MI455X (gfx1250) — compile-verified
//
#include <hip/hip_runtime.h>

typedef float v2f __attribute__((ext_vector_type(2)));
typedef float v8f __attribute__((ext_vector_type(8)));

#define DFEAT 128

// ---------------------------------------------------------------------------
// Zero a u32 array (deg / cursor must be zeroed every call; ws is poisoned).
// ---------------------------------------------------------------------------
__global__ void zero_u32_kernel(unsigned* __restrict__ p, int n) {
    int i = blockIdx.x * blockDim.x + threadIdx.x;
    if (i < n) p[i] = 0u;
}

// ---------------------------------------------------------------------------
// CSR build step 1: per-destination-row degree histogram (u32 atomics, cheap).
// ---------------------------------------------------------------------------
__global__ void degree_kernel(const int* __restrict__ erow,
                              unsigned* __restrict__ deg, int ne) {
    int i = blockIdx.x * blockDim.x + threadIdx.x;
    if (i < ne) atomicAdd(&deg[erow[i]], 1u);
}

// ---------------------------------------------------------------------------
// CSR build step 2: exclusive scan of deg -> rowptr (single 1024-thread block,
// N=100k -> 98 elems/thread; serial 1024-way combine in thread 0 is ~us).
// ---------------------------------------------------------------------------
__global__ __launch_bounds__(1024)
void scan_kernel(const unsigned* __restrict__ deg,
                 unsigned* __restrict__ rowptr, int n) {
    __shared__ unsigned ssum[1024];
    const int tid = threadIdx.x;
    const int ch  = (n + 1023) / 1024;
    const int lo  = tid * ch;
    const int hi  = (lo + ch < n) ? (lo + ch) : n;

    unsigned s = 0u;
    for (int i = lo; i < hi; ++i) s += deg[i];
    ssum[tid] = s;
    __syncthreads();

    if (tid == 0) {
        unsigned run = 0u;
        for (int i = 0; i < 1024; ++i) { unsigned t = ssum[i]; ssum[i] = run; run += t; }
        rowptr[n] = run;  // total edge count
    }
    __syncthreads();

    unsigned run = ssum[tid];
    for (int i = lo; i < hi; ++i) { rowptr[i] = run; run += deg[i]; }
}

// ---------------------------------------------------------------------------
// CSR build step 3: scatter edges into CSR slots.
// ---------------------------------------------------------------------------
__global__ void fill_kernel(const int* __restrict__ erow,
                            const int* __restrict__ ecol,
                            const float* __restrict__ eval,
                            const unsigned* __restrict__ rowptr,
                            unsigned* __restrict__ cursor,
                            int* __restrict__ csr_col,
                            float* __restrict__ csr_val, int ne) {
    int i = blockIdx.x * blockDim.x + threadIdx.x;
    if (i >= ne) return;
    int r = erow[i];
    unsigned p = atomicAdd(&cursor[r], 1u);
    unsigned idx = rowptr[r] + p;
    csr_col[idx] = ecol[i];
    csr_val[idx] = eval[i];
}

// ---------------------------------------------------------------------------
// GEMM: support = x @ W using V_WMMA_F32_16X16X4_F32 (exact fp32).
// Block = 256 threads = 8 waves; block b owns M-tile rows [16b,16b+16);
// wave w (0..7) owns N-tile w (D_OUT=128 = 8 tiles of 16). K loop: 32 x k=4.
// W (64KB) + x-tile (8KB) staged in LDS, rows padded to 129 floats (no bank
// conflicts: 129 % 64 != 0).
// ---------------------------------------------------------------------------
__global__ __launch_bounds__(256)
void gemm_wmma_kernel(const float* __restrict__ x,
                      const float* __restrict__ w,
                      float* __restrict__ support, int n) {
    __shared__ float lw[DFEAT * 129];   // W[k][nc] at lw[k*129 + nc]
    __shared__ float lx[16 * 129];      // x-tile[m][k] at lx[m*129 + k]

    const int tid  = threadIdx.x;
    const int wave = tid >> 5;          // 0..7 -> N tile
    const int lane = tid & 31;
    const int m    = lane & 15;
    const int half = lane >> 4;         // 0 or 1
    const int rowBase = blockIdx.x * 16;

    // Stage W: 128 rows x 32 float4 = 4096 float4, 16 per thread.
    for (int i = tid; i < DFEAT * 32; i += 256) {
        int r = i >> 5, c4 = i & 31;
        float4 v = ((const float4*)w)[r * 32 + c4];
        float* dst = &lw[r * 129 + c4 * 4];
        dst[0] = v.x; dst[1] = v.y; dst[2] = v.z; dst[3] = v.w;
    }
    // Stage x tile: 16 rows x 32 float4, 2 per thread (clamp keeps EXEC full).
    for (int i = tid; i < 16 * 32; i += 256) {
        int r = i >> 5, c4 = i & 31;
        int grow = rowBase + r; if (grow >= n) grow = n - 1;
        float4 v = ((const float4*)(x + (size_t)grow * DFEAT))[c4];
        float* dst = &lx[r * 129 + c4 * 4];
        dst[0] = v.x; dst[1] = v.y; dst[2] = v.z; dst[3] = v.w;
    }
    __syncthreads();

    // A 16x4 f32 layout: VGPR0 = K0(l0-15)/K2(l16-31), VGPR1 = K1/K3.
    // B 4x16 f32 layout mirrors it per-K-row across lanes (n = lane&15).
    const int nc = wave * 16 + m;
    v8f c = {};
#pragma unroll 8
    for (int kc = 0; kc < 32; ++kc) {
        const int kb = kc * 4 + 2 * half;
        v2f a, b;
        a.x = lx[m * 129 + kb];
        a.y = lx[m * 129 + kb + 1];
        b.x = lw[kb * 129 + nc];
        b.y = lw[(kb + 1) * 129 + nc];
        c = __builtin_amdgcn_wmma_f32_16x16x4_f32(
                /*neg_a=*/false, a, /*neg_b=*/false, b,
                /*c_mod=*/(short)0, c, /*reuse_a=*/false, /*reuse_b=*/false);
    }

    // C/D layout: VGPR r -> M=r (lanes 0-15), M=r+8 (lanes 16-31); N = lane&15.
    // Tile-in-range test is uniform per block -> one scalar branch, and the
    // common path (N % 16 == 0) stores unguarded (no per-store EXEC juggling).
    float* outp = support + (size_t)rowBase * DFEAT + wave * 16 + m + (size_t)half * 8 * DFEAT;
    if (rowBase + 16 <= n) {
#pragma unroll
        for (int r = 0; r < 8; ++r) outp[(size_t)r * DFEAT] = c[r];
    } else {
#pragma unroll
        for (int r = 0; r < 8; ++r) {
            if (rowBase + r + 8 * half < n) outp[(size_t)r * DFEAT] = c[r];
        }
    }
}

// ---------------------------------------------------------------------------
// Gather-SpMM + bias: one wave per destination row; lane owns 4 output floats
// in registers; (col,val) broadcast via v_readlane (uniform lane index) so
// they live in SGPRs: gather address = SGPR base + lane offset, scale operand
// is scalar. Zero fp atomics; all gathers hit L2 (support 51.2MB << 192MB).
// ---------------------------------------------------------------------------
__global__ __launch_bounds__(256)
void spmm_kernel(const float* __restrict__ support,
                 const unsigned* __restrict__ rowptr,
                 const int* __restrict__ csr_col,
                 const float* __restrict__ csr_val,
                 const float* __restrict__ bias,
                 float* __restrict__ out, int n) {
    const int lane = threadIdx.x & 31;
    const int row  = blockIdx.x * 8 + (threadIdx.x >> 5);
    if (row >= n) return;

    const float4 b4 = ((const float4*)bias)[lane];
    float4 acc = b4;                       // out = segsum + bias (fused init)

    const unsigned s = rowptr[row];
    const unsigned e = rowptr[row + 1];
    for (unsigned base = s; base < e; base += 32) {
        const unsigned i = base + lane;
        int cc = 0, vv = 0;
        if (i < e) {
            cc = csr_col[i];
            vv = ((const int*)csr_val)[i];
        }
        const int cnt = (int)((e - base < 32u) ? (e - base) : 32u);
        for (int j = 0; j < cnt; ++j) {
            const int   col = __builtin_amdgcn_readlane(cc, j);   // SGPR
            const int   vbits = __builtin_amdgcn_readlane(vv, j); // SGPR
            const float v = __builtin_bit_cast(float, vbits);
            const float4 srow = ((const float4*)(support + (size_t)col * DFEAT))[lane];
            acc.x += srow.x * v;
            acc.y += srow.y * v;
            acc.z += srow.z * v;
            acc.w += srow.w * v;
        }
    }
    ((float4*)(out + (size_t)row * DFEAT))[lane] = acc;
}

// ---------------------------------------------------------------------------
static inline size_t alignUp(size_t v, size_t a) { return (v + a - 1) & ~(a - 1); }

extern "C" void kernel_launch(void* const* d_in, const int* in_sizes, int n_in,
                              void* d_out, int out_size, void* d_ws, size_t ws_size,
                              hipStream_t stream) {
    const float* x     = (const float*)d_in[0];
    const int*   erow  = (const int*)  d_in[1];
    const int*   ecol  = (const int*)  d_in[2];
    const float* eval  = (const float*)d_in[3];
    const float* w     = (const float*)d_in[4];
    const float* bias  = (const float*)d_in[5];
    float*       out   = (float*)d_out;

    const int n  = in_sizes[0] / DFEAT;   // 100000
    const int ne = in_sizes[1];           // 3200000

    // Workspace carve-up (~78 MB total).
    char* ws = (char*)d_ws;
    size_t off = 0;
    float*    support = (float*)(ws + off);  off = alignUp(off + (size_t)n * DFEAT * 4, 256);
    unsigned* deg     = (unsigned*)(ws + off); off = alignUp(off + (size_t)n * 4, 256);
    unsigned* rowptr  = (unsigned*)(ws + off); off = alignUp(off + ((size_t)n + 1) * 4, 256);
    unsigned* cursor  = (unsigned*)(ws + off); off = alignUp(off + (size_t)n * 4, 256);
    int*      csr_col = (int*)(ws + off);    off = alignUp(off + (size_t)ne * 4, 256);
    float*    csr_val = (float*)(ws + off);  off = alignUp(off + (size_t)ne * 4, 256);
    (void)ws_size;

    // 1) zero histogram + cursors
    zero_u32_kernel<<<(n + 255) / 256, 256, 0, stream>>>(deg, n);
    zero_u32_kernel<<<(n + 255) / 256, 256, 0, stream>>>(cursor, n);

    // 2) GEMM support = x @ W (WMMA f32) — independent of CSR build, overlaps.
    gemm_wmma_kernel<<<(n + 15) / 16, 256, 0, stream>>>(x, w, support, n);

    // 3) CSR build
    degree_kernel<<<(ne + 255) / 256, 256, 0, stream>>>(erow, deg, ne);
    scan_kernel<<<1, 1024, 0, stream>>>(deg, rowptr, n);
    fill_kernel<<<(ne + 255) / 256, 256, 0, stream>>>(erow, ecol, eval, rowptr,
                                                      cursor, csr_col, csr_val, ne);

    // 4) gather-SpMM + bias
    spmm_kernel<<<(n + 7) / 8, 256, 0, stream>>>(support, rowptr, csr_col,
                                                 csr_val, bias, out, n);
}